// Sobel3d_78374563218011
// MI455X (gfx1250) — compile-verified
//
#include <hip/hip_runtime.h>
#include <stdint.h>

#define DSZ 192
#define TX 32
#define TY 8
#define HTX 34
#define HTY 10
#define TILE (HTX * HTY)     // 340 floats per input per plane
#define ZC 48                // z-chunk per block
#define NBLK ((DSZ / TX) * (DSZ / TY) * 2 * (DSZ / ZC))   // 6*24*8 = 1152

// ---- CDNA5 async global->LDS path (guarded; fallback = load + ds_store) ----
#if defined(__has_builtin)
#if __has_builtin(__builtin_amdgcn_global_load_async_to_lds_b32) && \
    __has_builtin(__builtin_amdgcn_s_wait_asynccnt)
#define USE_ASYNC 1
#endif
#endif
#ifndef USE_ASYNC
#define USE_ASYNC 0
#endif

#if USE_ASYNC
#define WAIT_ASYNC() __builtin_amdgcn_s_wait_asynccnt(0)
// builtin signature wants int* in AS1/AS3 (clang prints them as '__device__ int *')
#define AS1P(p) ((__attribute__((address_space(1))) int*)(unsigned long long)(uintptr_t)(p))
// generic pointer to LDS carries the LDS byte offset in its low 32 bits
#define AS3P(p) ((__attribute__((address_space(3))) int*)(unsigned int)(uintptr_t)(p))
#else
#define WAIT_ASYNC()
#endif

// Issue one halo plane (34x10) of input `base` at depth zp into LDS buf.
// OOB cells (zero padding) are written as 0.0f directly.
__device__ __forceinline__ void issue_plane(const float* __restrict__ base, int zp,
                                            float* buf, int gx0, int gy0, int tid) {
    const bool zin = (zp >= 0) & (zp < DSZ);
#pragma unroll
    for (int i = tid; i < TILE; i += TX * TY) {
        const int hy = i / HTX;
        const int hx = i - hy * HTX;
        const int gx = gx0 + hx - 1;
        const int gy = gy0 + hy - 1;
        const bool in = zin & (gx >= 0) & (gx < DSZ) & (gy >= 0) & (gy < DSZ);
        if (in) {
            const float* g = base + (((size_t)zp * DSZ + gy) * DSZ + gx);
#if USE_ASYNC
            __builtin_amdgcn_global_load_async_to_lds_b32(AS1P(g), AS3P(buf + i), 0, 0);
#else
            buf[i] = *g;
#endif
        } else {
            buf[i] = 0.0f;
        }
    }
}

// From a halo tile, compute the three separable intermediates at (lx,ly):
//   a = Sy(Sx v), b = Sy(Dx v), c = Dy(Sx v)
__device__ __forceinline__ void abc_from_tile(const float* buf, int lx, int ly,
                                              float& a, float& b, float& c) {
    const float* r0 = buf + (ly + 0) * HTX + lx;
    const float* r1 = buf + (ly + 1) * HTX + lx;
    const float* r2 = buf + (ly + 2) * HTX + lx;
    const float s0 = r0[0] + 2.0f * r0[1] + r0[2];
    const float d0 = r0[2] - r0[0];
    const float s1 = r1[0] + 2.0f * r1[1] + r1[2];
    const float d1 = r1[2] - r1[0];
    const float s2 = r2[0] + 2.0f * r2[1] + r2[2];
    const float d2 = r2[2] - r2[0];
    a = s0 + 2.0f * s1 + s2;
    b = d0 + 2.0f * d1 + d2;
    c = s2 - s0;
}

__global__ __launch_bounds__(TX * TY)
void sobel3d_partial(const float* __restrict__ A, const float* __restrict__ B,
                     float* __restrict__ partial) {
    __shared__ float tiles[2][2][TILE];   // [parity][input]
    __shared__ float red[TX * TY];

    const int lx  = threadIdx.x;
    const int ly  = threadIdx.y;
    const int tid = ly * TX + lx;
    const int gx0 = blockIdx.x * TX;
    const int gy0 = blockIdx.y * TY;
    const int batch = (int)blockIdx.z >> 2;
    const int chunk = (int)blockIdx.z & 3;
    const int z0 = chunk * ZC;
    const int z1 = z0 + ZC;

    const float* baseA = A + (size_t)batch * DSZ * DSZ * DSZ;
    const float* baseB = B + (size_t)batch * DSZ * DSZ * DSZ;

    float aA[3], bA[3], cA[3];
    float aB[3], bB[3], cB[3];

    // ---- prologue: planes z0-1 (parity 0) and z0 (parity 1) ----
    issue_plane(baseA, z0 - 1, tiles[0][0], gx0, gy0, tid);
    issue_plane(baseB, z0 - 1, tiles[0][1], gx0, gy0, tid);
    WAIT_ASYNC();
    __syncthreads();

    issue_plane(baseA, z0, tiles[1][0], gx0, gy0, tid);
    issue_plane(baseB, z0, tiles[1][1], gx0, gy0, tid);
    {
        const int s = z0 % 3;   // slot for plane z0-1
        abc_from_tile(tiles[0][0], lx, ly, aA[s], bA[s], cA[s]);
        abc_from_tile(tiles[0][1], lx, ly, aB[s], bB[s], cB[s]);
    }
    WAIT_ASYNC();
    __syncthreads();

    issue_plane(baseA, z0 + 1, tiles[0][0], gx0, gy0, tid);
    issue_plane(baseB, z0 + 1, tiles[0][1], gx0, gy0, tid);
    {
        const int s = (z0 + 1) % 3;   // slot for plane z0
        abc_from_tile(tiles[1][0], lx, ly, aA[s], bA[s], cA[s]);
        abc_from_tile(tiles[1][1], lx, ly, aB[s], bB[s], cB[s]);
    }

    // ---- main loop over output planes z in [z0, z1) ----
    float acc = 0.0f;
    for (int z = z0; z < z1; ++z) {
        WAIT_ASYNC();
        __syncthreads();                 // plane z+1 is now resident in parity pr
        const int i  = z - z0;
        const int pr = i & 1;

        if (z + 2 <= z1) {               // stage plane z+2 into the other parity
            issue_plane(baseA, z + 2, tiles[pr ^ 1][0], gx0, gy0, tid);
            issue_plane(baseB, z + 2, tiles[pr ^ 1][1], gx0, gy0, tid);
        }

        const int sn = (z + 2) % 3;      // slot for plane z+1
        abc_from_tile(tiles[pr][0], lx, ly, aA[sn], bA[sn], cA[sn]);
        abc_from_tile(tiles[pr][1], lx, ly, aB[sn], bB[sn], cB[sn]);

        const int sp = z % 3;            // plane z-1
        const int sc = (z + 1) % 3;      // plane z

        const float gx1 = bA[sp] + 2.0f * bA[sc] + bA[sn];
        const float gy1 = cA[sp] + 2.0f * cA[sc] + cA[sn];
        const float gz1 = aA[sn] - aA[sp];
        const float G1  = fabsf(gx1) + fabsf(gy1) + fabsf(gz1);

        const float gx2 = bB[sp] + 2.0f * bB[sc] + bB[sn];
        const float gy2 = cB[sp] + 2.0f * cB[sc] + cB[sn];
        const float gz2 = aB[sn] - aB[sp];
        const float G2  = fabsf(gx2) + fabsf(gy2) + fabsf(gz2);

        acc += fabsf(G1 - G2);
    }

    // ---- deterministic block reduction ----
    red[tid] = acc;
    __syncthreads();
#pragma unroll
    for (int k = (TX * TY) / 2; k > 0; k >>= 1) {
        if (tid < k) red[tid] += red[tid + k];
        __syncthreads();
    }
    if (tid == 0) {
        const int bid = (int)blockIdx.x +
                        (int)gridDim.x * ((int)blockIdx.y + (int)gridDim.y * (int)blockIdx.z);
        partial[bid] = red[0];
    }
}

__global__ __launch_bounds__(256)
void sobel3d_reduce(const float* __restrict__ partial, float* __restrict__ out, int n) {
    __shared__ double red[256];
    const int t = threadIdx.x;
    double s = 0.0;
    for (int i = t; i < n; i += 256) s += (double)partial[i];
    red[t] = s;
    __syncthreads();
#pragma unroll
    for (int k = 128; k > 0; k >>= 1) {
        if (t < k) red[t] += red[t + k];
        __syncthreads();
    }
    if (t == 0) out[0] = (float)(red[0] / (double)(2ull * DSZ * DSZ * DSZ));
}

extern "C" void kernel_launch(void* const* d_in, const int* in_sizes, int n_in,
                              void* d_out, int out_size, void* d_ws, size_t ws_size,
                              hipStream_t stream) {
    (void)in_sizes; (void)n_in; (void)out_size; (void)ws_size;
    const float* A = (const float*)d_in[0];
    const float* B = (const float*)d_in[1];
    float* partial = (float*)d_ws;

    dim3 grid(DSZ / TX, DSZ / TY, 2 * (DSZ / ZC));   // 6 x 24 x 8 = 1152 blocks
    dim3 block(TX, TY, 1);                            // 256 threads = 8 wave32
    sobel3d_partial<<<grid, block, 0, stream>>>(A, B, partial);
    sobel3d_reduce<<<1, 256, 0, stream>>>(partial, (float*)d_out, NBLK);
}